// MultiLayerTetra_36979668419174
// MI455X (gfx1250) — compile-verified
//
#include <hip/hip_runtime.h>

#define TETRA_MAX_LAYER 20

typedef __attribute__((ext_vector_type(2))) float v2f;
typedef __attribute__((ext_vector_type(8))) float v8f;

__device__ __forceinline__ float sel4(int c, float a0, float a1, float a2, float a3) {
  return (c == 0) ? a0 : (c == 1) ? a1 : (c == 2) ? a2 : a3;
}

// Barycentric coords of q in tetra (v0,v1,v2,v3) via Cramer on the edge matrix.
__device__ __forceinline__ void barycentric(
    float qx, float qy, float qz,
    float v0x, float v0y, float v0z,
    float v1x, float v1y, float v1z,
    float v2x, float v2y, float v2z,
    float v3x, float v3y, float v3z,
    float& b0, float& b1, float& b2, float& b3) {
  float e0x = v0x - v3x, e0y = v0y - v3y, e0z = v0z - v3z;
  float e1x = v1x - v3x, e1y = v1y - v3y, e1z = v1z - v3z;
  float e2x = v2x - v3x, e2y = v2y - v3y, e2z = v2z - v3z;
  float rx = qx - v3x, ry = qy - v3y, rz = qz - v3z;
  float cxx = e1y * e2z - e1z * e2y;           // e1 x e2
  float cxy = e1z * e2x - e1x * e2z;
  float cxz = e1x * e2y - e1y * e2x;
  float det = e0x * cxx + e0y * cxy + e0z * cxz;
  float inv = 1.0f / det;
  float n0 = rx * cxx + ry * cxy + rz * cxz;   // det(r,e1,e2)
  float rxx = ry * e2z - rz * e2y;             // r x e2
  float rxy = rz * e2x - rx * e2z;
  float rxz = rx * e2y - ry * e2x;
  float n1 = e0x * rxx + e0y * rxy + e0z * rxz; // det(e0,r,e2)
  float exx = e1y * rz - e1z * ry;             // e1 x r
  float exy = e1z * rx - e1x * rz;
  float exz = e1x * ry - e1y * rx;
  float n2 = e0x * exx + e0y * exy + e0z * exz; // det(e0,e1,r)
  b0 = n0 * inv;
  b1 = n1 * inv;
  b2 = n2 * inv;
  b3 = 1.0f - b0 - b1 - b2;
}

// One thread per point: walk the tree carrying compressed state
// (12 vertex floats + 4 feature-row indices). Emits {can[4], fidx[4]} to stash.
__global__ void tetra_traverse(const float* __restrict__ xyz,
                               const float* __restrict__ root_xyz,
                               const int* __restrict__ child_index,
                               const int* __restrict__ point_index,
                               const int* __restrict__ child_cut,
                               const int* __restrict__ act_layer,
                               const float* __restrict__ field,
                               float* __restrict__ stash, int stride,
                               int n, int F) {
  int p = blockIdx.x * blockDim.x + threadIdx.x;
  if (p >= n) return;

  float qx = xyz[p * 3 + 0], qy = xyz[p * 3 + 1], qz = xyz[p * 3 + 2];
  float v0x = root_xyz[0],  v0y = root_xyz[1],  v0z = root_xyz[2];
  float v1x = root_xyz[3],  v1y = root_xyz[4],  v1z = root_xyz[5];
  float v2x = root_xyz[6],  v2y = root_xyz[7],  v2z = root_xyz[8];
  float v3x = root_xyz[9],  v3y = root_xyz[10], v3z = root_xyz[11];

  int f0 = 0, f1 = 1, f2 = 2, f3 = 3;       // cfeat rows == field[point_index[0]] == field[0:4]
  int c0 = child_cut[0], c1 = child_cut[1];
  int cact = act_layer[0];
  int cell = 0;
  float b0, b1, b2, b3;

  for (int i = 1; i <= TETRA_MAX_LAYER - 2; ++i) {
    int act = (cact < TETRA_MAX_LAYER) ? cact : TETRA_MAX_LAYER;
    if (act != i) continue;                  // state only changes when act == i
    barycentric(qx, qy, qz, v0x, v0y, v0z, v1x, v1y, v1z,
                v2x, v2y, v2z, v3x, v3y, v3z, b0, b1, b2, b3);
    float cv0 = sel4(c0, b0, b1, b2, b3);
    float cv1 = sel4(c1, b0, b1, b2, b3);
    int choice = (cv0 - cv1 > 0.0f) ? 1 : 0;
    int child = child_index[cell * 2 + choice];
    if (child == -1) continue;
    int ab = choice ? c0 : c1;               // ccut[1 - choice]
    int pi = point_index[child * 4 + ab];
    float mx = 0.5f * (sel4(c0, v0x, v1x, v2x, v3x) + sel4(c1, v0x, v1x, v2x, v3x));
    float my = 0.5f * (sel4(c0, v0y, v1y, v2y, v3y) + sel4(c1, v0y, v1y, v2y, v3y));
    float mz = 0.5f * (sel4(c0, v0z, v1z, v2z, v3z) + sel4(c1, v0z, v1z, v2z, v3z));
    v0x = (ab == 0) ? mx : v0x;  v0y = (ab == 0) ? my : v0y;  v0z = (ab == 0) ? mz : v0z;
    v1x = (ab == 1) ? mx : v1x;  v1y = (ab == 1) ? my : v1y;  v1z = (ab == 1) ? mz : v1z;
    v2x = (ab == 2) ? mx : v2x;  v2y = (ab == 2) ? my : v2y;  v2z = (ab == 2) ? mz : v2z;
    v3x = (ab == 3) ? mx : v3x;  v3y = (ab == 3) ? my : v3y;  v3z = (ab == 3) ? mz : v3z;
    f0 = (ab == 0) ? pi : f0;  f1 = (ab == 1) ? pi : f1;
    f2 = (ab == 2) ? pi : f2;  f3 = (ab == 3) ? pi : f3;
    c0 = child_cut[child * 2];  c1 = child_cut[child * 2 + 1];
    cact = act_layer[child];
    cell = child;
  }

  barycentric(qx, qy, qz, v0x, v0y, v0z, v1x, v1y, v1z,
              v2x, v2y, v2z, v3x, v3y, v3z, b0, b1, b2, b3);

  // Warm L2 (192MB easily holds field) for the combine pass: global_prefetch_b8.
  __builtin_prefetch(field + (size_t)f0 * F, 0, 1);
  __builtin_prefetch(field + (size_t)f1 * F, 0, 1);
  __builtin_prefetch(field + (size_t)f2 * F, 0, 1);
  __builtin_prefetch(field + (size_t)f3 * F, 0, 1);

  float* sp = stash + (size_t)p * stride;
  *(float4*)(sp) = make_float4(b0, b1, b2, b3);
  *(int4*)(sp + 4) = make_int4(f0, f1, f2, f3);
}

// One wave == 16 points. Out(16x32) = blockdiag(can_p) x stacked field rows,
// expressed as 16 accumulator-chained V_WMMA_F32_16X16X4_F32 per feature half.
// A 16x4 layout: V0 = K0 (lanes0-15) / K2 (lanes16-31), V1 = K1 / K3.
// B 4x16 layout: V0 = K0 / K2, V1 = K1 / K3, N = lane%16.
// D 16x16 layout: Vj = M=j (lanes0-15) / M=j+8 (lanes16-31), N = lane%16.
__global__ void tetra_combine_wmma(const float* __restrict__ stash, int stride,
                                   const float* __restrict__ field,
                                   float* __restrict__ out, int groups) {
  int lane = threadIdx.x & 31;
  int wave = threadIdx.x >> 5;
  int group = blockIdx.x * (blockDim.x >> 5) + wave;
  if (group >= groups) return;             // wave-uniform: EXEC stays all-ones for WMMA
  int base = group * 16;
  int l15 = lane & 15;
  bool hi = lane >= 16;

  // Both lane halves preload point (base + l15)'s stash: can + feature indices.
  const float* sp = stash + (size_t)(base + l15) * stride;
  float mc0 = sp[0], mc1 = sp[1], mc2 = sp[2], mc3 = sp[3];
  const int* ip = (const int*)(sp + 4);
  int mi0 = ip[0], mi1 = ip[1], mi2 = ip[2], mi3 = ip[3];

  float axv = hi ? mc2 : mc0;              // my A contribution (K0/K2, K1/K3)
  float ayv = hi ? mc3 : mc1;
  int r02s = hi ? mi2 : mi0;               // my B row ids (K0/K2, K1/K3)
  int r13s = hi ? mi3 : mi1;

  v8f acc0 = {0, 0, 0, 0, 0, 0, 0, 0};     // features 0..15
  v8f acc1 = {0, 0, 0, 0, 0, 0, 0, 0};     // features 16..31

  for (int p = 0; p < 16; ++p) {
    bool mine = (l15 == p);
    v2f a;
    a.x = mine ? axv : 0.0f;               // A chunk p: only row p nonzero
    a.y = mine ? ayv : 0.0f;
    int src = hi ? (p + 16) : p;           // low half sources fidx[0,1], high half fidx[2,3]
    int r02 = __shfl(r02s, src, 32);
    int r13 = __shfl(r13s, src, 32);
    const float* row02 = field + (size_t)r02 * 32;
    const float* row13 = field + (size_t)r13 * 32;
    v2f b0, b1;
    b0.x = row02[l15];       b0.y = row13[l15];
    b1.x = row02[16 + l15];  b1.y = row13[16 + l15];
    acc0 = __builtin_amdgcn_wmma_f32_16x16x4_f32(false, a, false, b0,
                                                 (short)0, acc0, false, false);
    acc1 = __builtin_amdgcn_wmma_f32_16x16x4_f32(false, a, false, b1,
                                                 (short)0, acc1, false, false);
  }

  int rbase = base + (hi ? 8 : 0);
#pragma unroll
  for (int j = 0; j < 8; ++j) {
    float* orow = out + (size_t)(rbase + j) * 32;
    orow[l15] = acc0[j];
    orow[16 + l15] = acc1[j];
  }
}

// Generic fallback (tail points / F != 32). One thread per point; stash is
// fully register-held before any store, so it is safe even when stash == out.
__global__ void tetra_combine_generic(const float* __restrict__ stash, int stride,
                                      const float* __restrict__ field,
                                      float* __restrict__ out,
                                      int start, int n, int F) {
  int p = start + blockIdx.x * blockDim.x + threadIdx.x;
  if (p >= n) return;
  const float* sp = stash + (size_t)p * stride;
  float c0 = sp[0], c1 = sp[1], c2 = sp[2], c3 = sp[3];
  const int* ip = (const int*)(sp + 4);
  int i0 = ip[0], i1 = ip[1], i2 = ip[2], i3 = ip[3];
  const float* r0 = field + (size_t)i0 * F;
  const float* r1 = field + (size_t)i1 * F;
  const float* r2 = field + (size_t)i2 * F;
  const float* r3 = field + (size_t)i3 * F;
  for (int f = 0; f < F; ++f)
    out[(size_t)p * F + f] = c0 * r0[f] + c1 * r1[f] + c2 * r2[f] + c3 * r3[f];
}

extern "C" void kernel_launch(void* const* d_in, const int* in_sizes, int n_in,
                              void* d_out, int out_size, void* d_ws, size_t ws_size,
                              hipStream_t stream) {
  (void)n_in;
  const float* xyz         = (const float*)d_in[0];
  const float* field       = (const float*)d_in[1];
  const float* root        = (const float*)d_in[2];
  const int*   child_index = (const int*)d_in[3];
  const int*   point_index = (const int*)d_in[4];
  const int*   child_cut   = (const int*)d_in[5];
  const int*   act_layer   = (const int*)d_in[6];
  float* out = (float*)d_out;

  int n = in_sizes[0] / 3;
  int F = (n > 0) ? (out_size / n) : 0;

  // Stash layout per point: {can[4] f32, fidx[4] i32} = 32B.
  size_t need = (size_t)n * 8 * sizeof(float);
  float* stash;
  int stride;
  if (ws_size >= need) { stash = (float*)d_ws; stride = 8; }
  else                 { stash = out;          stride = F; }  // stage inside out rows (F>=8)

  dim3 blk(256);
  dim3 grd((unsigned)((n + 255) / 256));
  tetra_traverse<<<grd, blk, 0, stream>>>(xyz, root, child_index, point_index,
                                          child_cut, act_layer, field,
                                          stash, stride, n, F);

  if (F == 32) {
    int groups = n / 16;
    if (groups > 0) {
      int blocks = (groups + 7) / 8;       // 8 wave32 per 256-thread block
      tetra_combine_wmma<<<blocks, 256, 0, stream>>>(stash, stride, field, out, groups);
    }
    int rem = n - groups * 16;
    if (rem > 0)
      tetra_combine_generic<<<(rem + 255) / 256, 256, 0, stream>>>(
          stash, stride, field, out, groups * 16, n, F);
  } else if (n > 0) {
    tetra_combine_generic<<<(n + 255) / 256, 256, 0, stream>>>(
        stash, stride, field, out, 0, n, F);
  }
}